// channelMix_CI_pat_78417512890550
// MI455X (gfx1250) — compile-verified
//
#include <hip/hip_runtime.h>
#include <math.h>

typedef __attribute__((ext_vector_type(8))) float v8f;
typedef __attribute__((ext_vector_type(2))) float v2f;

#define B_TOT 1024
#define P     84
#define CDIM  512
#define BT    128      // batch rows per tile
#define XS    85       // X LDS row stride (odd -> conflict-free)
#define HS    97       // H LDS row stride (odd, >= 96 cols)
#define WS    85       // W LDS row stride
#define QPAD  96       // q dimension padded to 6 WMMA tiles
#define KU    3        // k-step unroll (3 x 4 = 12, 84 = 7*12 exact)

__device__ __forceinline__ float gelu_exact(float v) {
    return 0.5f * v * (1.0f + erff(v * 0.70710678118654752f));
}

__global__ void __launch_bounds__(256)
mix_kernel(const float* __restrict__ x,
           const float* __restrict__ W1, const float* __restrict__ b1,
           const float* __restrict__ W2, const float* __restrict__ b2,
           float* __restrict__ out) {
    extern __shared__ float smem[];
    float* Xs  = smem;                 // BT*XS
    float* Hsm = Xs  + BT * XS;        // BT*HS
    float* W1s = Hsm + BT * HS;        // QPAD*WS
    float* W2s = W1s + QPAD * WS;      // QPAD*WS
    float* b1s = W2s + QPAD * WS;      // QPAD
    float* b2s = b1s + QPAD;           // QPAD

    const int c    = blockIdx.x;       // channel
    const int tid  = threadIdx.x;
    const int lane = tid & 31;
    const int w    = tid >> 5;         // wave id 0..7
    const int l15  = lane & 15;
    const int half = lane >> 4;        // 0 or 1
    const int m0   = w * 16;           // this wave's M tile

    // ---- zero padded weight/bias regions, then load this channel's weights
    for (int i = tid; i < QPAD * WS; i += 256) { W1s[i] = 0.0f; W2s[i] = 0.0f; }
    for (int i = tid; i < QPAD; i += 256)      { b1s[i] = 0.0f; b2s[i] = 0.0f; }
    __syncthreads();
    const float* W1c = W1 + (size_t)c * P * P;
    const float* W2c = W2 + (size_t)c * P * P;
    for (int i = tid; i < P * P; i += 256) {
        int r = i / P, col = i - r * P;
        W1s[r * WS + col] = W1c[i];
        W2s[r * WS + col] = W2c[i];
    }
    for (int i = tid; i < P; i += 256) {
        b1s[i] = b1[(size_t)c * P + i];
        b2s[i] = b2[(size_t)c * P + i];
    }

    for (int b0 = 0; b0 < B_TOT; b0 += BT) {
        __syncthreads();   // weights visible / previous tile fully consumed
        // ---- stage X tile: x[b0+bb, pp, c]  (L2 absorbs cross-channel reuse)
        for (int i = tid; i < BT * P; i += 256) {
            int bb = i / P, pp = i - bb * P;
            Xs[bb * XS + pp] = x[((size_t)(b0 + bb) * P + pp) * CDIM + c];
        }
        __syncthreads();

        // ================= GEMM1: H = X @ W1^T =================
        v8f acc[6] = {};
        for (int k = 0; k < P; k += 4 * KU) {
            v2f a[KU], bm[KU][6];
#pragma unroll
            for (int u = 0; u < KU; ++u) {
                const int acol = k + 4 * u + 2 * half;
                a[u].x = Xs[(m0 + l15) * XS + acol];
                a[u].y = Xs[(m0 + l15) * XS + acol + 1];
#pragma unroll
                for (int n = 0; n < 6; ++n) {
                    bm[u][n].x = W1s[(n * 16 + l15) * WS + acol];
                    bm[u][n].y = W1s[(n * 16 + l15) * WS + acol + 1];
                }
            }
#pragma unroll
            for (int u = 0; u < KU; ++u)
#pragma unroll
                for (int n = 0; n < 6; ++n)
                    acc[n] = __builtin_amdgcn_wmma_f32_16x16x4_f32(
                        false, a[u], false, bm[u][n], (short)0, acc[n], false, false);
        }
        // bias + exact GELU -> H in LDS
#pragma unroll
        for (int n = 0; n < 6; ++n) {
            const int q = n * 16 + l15;
            const float bias = b1s[q];
#pragma unroll
            for (int r = 0; r < 8; ++r) {
                const int m = m0 + r + 8 * half;
                Hsm[m * HS + q] = gelu_exact(acc[n][r] + bias);
            }
        }
        __syncthreads();

        // ================= GEMM2: O = H @ W2^T =================
#pragma unroll
        for (int n = 0; n < 6; ++n) acc[n] = (v8f){0,0,0,0,0,0,0,0};
        for (int k = 0; k < P; k += 4 * KU) {
            v2f a[KU], bm[KU][6];
#pragma unroll
            for (int u = 0; u < KU; ++u) {
                const int acol = k + 4 * u + 2 * half;
                a[u].x = Hsm[(m0 + l15) * HS + acol];
                a[u].y = Hsm[(m0 + l15) * HS + acol + 1];
#pragma unroll
                for (int n = 0; n < 6; ++n) {
                    bm[u][n].x = W2s[(n * 16 + l15) * WS + acol];
                    bm[u][n].y = W2s[(n * 16 + l15) * WS + acol + 1];
                }
            }
#pragma unroll
            for (int u = 0; u < KU; ++u)
#pragma unroll
                for (int n = 0; n < 6; ++n)
                    acc[n] = __builtin_amdgcn_wmma_f32_16x16x4_f32(
                        false, a[u], false, bm[u][n], (short)0, acc[n], false, false);
        }
        // bias + residual, store pre-LN result
#pragma unroll
        for (int n = 0; n < 6; ++n) {
            const int q = n * 16 + l15;
            if (q < P) {
                const float bias = b2s[q];
#pragma unroll
                for (int r = 0; r < 8; ++r) {
                    const int m = m0 + r + 8 * half;
                    const float res = acc[n][r] + bias + Xs[m * XS + q];
                    out[((size_t)(b0 + m) * P + q) * CDIM + c] = res;
                }
            }
        }
    }
}

// In-place LayerNorm over last dim (512), one wave per row.
__global__ void __launch_bounds__(256)
ln_kernel(float* __restrict__ out,
          const float* __restrict__ gamma, const float* __restrict__ beta) {
    const int rows = B_TOT * P;
    const int gw = (int)((blockIdx.x * blockDim.x + threadIdx.x) >> 5);
    if (gw >= rows) return;
    const int lane = threadIdx.x & 31;
    float* row = out + (size_t)gw * CDIM;

    float4 v[4];
    float s = 0.0f;
#pragma unroll
    for (int j = 0; j < 4; ++j) {
        v[j] = ((const float4*)row)[lane + 32 * j];
        s += v[j].x + v[j].y + v[j].z + v[j].w;
    }
#pragma unroll
    for (int o = 16; o > 0; o >>= 1) s += __shfl_xor(s, o, 32);
    const float mu = s * (1.0f / CDIM);

    float vs = 0.0f;
#pragma unroll
    for (int j = 0; j < 4; ++j) {
        float dx = v[j].x - mu, dy = v[j].y - mu, dz = v[j].z - mu, dw = v[j].w - mu;
        vs += dx * dx + dy * dy + dz * dz + dw * dw;
    }
#pragma unroll
    for (int o = 16; o > 0; o >>= 1) vs += __shfl_xor(vs, o, 32);
    const float rstd = rsqrtf(vs * (1.0f / CDIM) + 1e-5f);

#pragma unroll
    for (int j = 0; j < 4; ++j) {
        const int idx = lane + 32 * j;
        float4 g  = ((const float4*)gamma)[idx];
        float4 be = ((const float4*)beta)[idx];
        float4 o4;
        o4.x = (v[j].x - mu) * rstd * g.x + be.x;
        o4.y = (v[j].y - mu) * rstd * g.y + be.y;
        o4.z = (v[j].z - mu) * rstd * g.z + be.z;
        o4.w = (v[j].w - mu) * rstd * g.w + be.w;
        ((float4*)row)[idx] = o4;
    }
}

extern "C" void kernel_launch(void* const* d_in, const int* in_sizes, int n_in,
                              void* d_out, int out_size, void* d_ws, size_t ws_size,
                              hipStream_t stream) {
    const float* x     = (const float*)d_in[0];
    const float* W1    = (const float*)d_in[1];
    const float* b1    = (const float*)d_in[2];
    const float* W2    = (const float*)d_in[3];
    const float* b2    = (const float*)d_in[4];
    const float* gamma = (const float*)d_in[5];
    const float* beta  = (const float*)d_in[6];
    float* out = (float*)d_out;

    const size_t lds_bytes =
        (size_t)(BT * XS + BT * HS + 2 * QPAD * WS + 2 * QPAD) * sizeof(float);
    mix_kernel<<<CDIM, 256, lds_bytes, stream>>>(x, W1, b1, W2, b2, out);

    const int rows = B_TOT * P;                    // 86016 rows, 1 wave each
    const int blocks = (rows * 32 + 255) / 256;    // 8 waves per block
    ln_kernel<<<blocks, 256, 0, stream>>>(out, gamma, beta);
}